// GNNPolicy_7713761264034
// MI455X (gfx1250) — compile-verified
//
#include <hip/hip_runtime.h>
#include <math.h>

typedef __attribute__((ext_vector_type(2))) float v2f;
typedef __attribute__((ext_vector_type(8))) float v8f;

// ---------------- init: deg=1 (self loop), zero accumulators & output ----------------
__global__ void k_init(float* deg, float* aggS, float4* a4a, float4* a4b,
                       float* out, int n) {
    int i = blockIdx.x * blockDim.x + threadIdx.x;
    if (i < 8) out[i] = 0.0f;
    if (i >= n) return;
    deg[i]  = 1.0f;           // self-loop contributes 1 to degree
    aggS[i] = 0.0f;
    a4a[i] = make_float4(0.f, 0.f, 0.f, 0.f);
    a4b[i] = make_float4(0.f, 0.f, 0.f, 0.f);
}

// ---------------- degree: count edges at target ----------------
__global__ void k_degree(const long long* __restrict__ col, float* __restrict__ deg, int e) {
    int i = blockIdx.x * blockDim.x + threadIdx.x;
    if (i >= e) return;
    atomicAdd(&deg[(int)col[i]], 1.0f);
}

// ---------------- dinv = 1/sqrt(deg), in place ----------------
__global__ void k_dinv(float* degdinv, int n) {
    int i = blockIdx.x * blockDim.x + threadIdx.x;
    if (i >= n) return;
    float d = degdinv[i];
    degdinv[i] = 1.0f / sqrtf(d);   // deg >= 1 always (self loop)
}

// ---------------- layer-1 edge aggregation (scalar, shared by both towers) ----------------
__global__ void k_agg1(const long long* __restrict__ row, const long long* __restrict__ col,
                       const float* __restrict__ dinv, const float* __restrict__ x,
                       float* __restrict__ aggS, int e) {
    int i = blockIdx.x * blockDim.x + threadIdx.x;
    if (i >= e) return;
    int r = (int)row[i], c = (int)col[i];
    atomicAdd(&aggS[c], dinv[r] * dinv[c] * x[r]);
}

// ---------------- node-wise layer 1: h = relu(agg * W + b), both towers ----------------
__global__ void k_layer1(const float* __restrict__ aggS, const float* __restrict__ dinv,
                         const float* __restrict__ x,
                         const float* __restrict__ w1, const float* __restrict__ b1,
                         const float* __restrict__ w2, const float* __restrict__ b2,
                         float4* __restrict__ h1, float4* __restrict__ h2, int n) {
    int i = blockIdx.x * blockDim.x + threadIdx.x;
    if (i >= n) return;
    float di = dinv[i];
    float a = aggS[i] + di * di * x[i];   // add self-loop message
    float4 o1, o2;
    o1.x = fmaxf(a * w1[0] + b1[0], 0.f);
    o1.y = fmaxf(a * w1[1] + b1[1], 0.f);
    o1.z = fmaxf(a * w1[2] + b1[2], 0.f);
    o1.w = fmaxf(a * w1[3] + b1[3], 0.f);
    o2.x = fmaxf(a * w2[0] + b2[0], 0.f);
    o2.y = fmaxf(a * w2[1] + b2[1], 0.f);
    o2.z = fmaxf(a * w2[2] + b2[2], 0.f);
    o2.w = fmaxf(a * w2[3] + b2[3], 0.f);
    h1[i] = o1;
    h2[i] = o2;
}

// ---------------- layer-2 edge aggregation (fused towers, 4-vectors) ----------------
__global__ void k_agg2(const long long* __restrict__ row, const long long* __restrict__ col,
                       const float* __restrict__ dinv,
                       const float4* __restrict__ h1, const float4* __restrict__ h2,
                       float* __restrict__ a4a, float* __restrict__ a4b, int e) {
    int i = blockIdx.x * blockDim.x + threadIdx.x;
    if (i >= e) return;
    int r = (int)row[i], c = (int)col[i];
    float nr = dinv[r] * dinv[c];
    float4 v1 = h1[r];
    float4 v2 = h2[r];
    float* pa = a4a + 4 * c;
    float* pb = a4b + 4 * c;
    atomicAdd(pa + 0, nr * v1.x);
    atomicAdd(pa + 1, nr * v1.y);
    atomicAdd(pa + 2, nr * v1.z);
    atomicAdd(pa + 3, nr * v1.w);
    atomicAdd(pb + 0, nr * v2.x);
    atomicAdd(pb + 1, nr * v2.y);
    atomicAdd(pb + 2, nr * v2.z);
    atomicAdd(pb + 3, nr * v2.w);
}

// ---------------- node-wise layer 2 via V_WMMA_F32_16X16X4_F32 ----------------
// Each wave handles 16 nodes. A(16x4) = agg4 (+self loop), B(4x16) = W zero-padded.
// A layout (32-bit, 16x4): lane m=l&15, K = vgpr + 2*(l>>4).
// B layout (4x16): lane n=l&15, row K = vgpr + 2*(l>>4).
// D layout (16x16 f32): lane col N=l&15, row M = vgpr + 8*(l>>4).
__global__ void k_layer2_wmma(const float* __restrict__ a4a, const float* __restrict__ a4b,
                              const float* __restrict__ h1, const float* __restrict__ h2,
                              const float* __restrict__ dinv,
                              const float* __restrict__ w1, const float* __restrict__ b1,
                              const float* __restrict__ w2, const float* __restrict__ b2,
                              float* __restrict__ y1, float* __restrict__ y2, int n) {
    int wave = (blockIdx.x * blockDim.x + threadIdx.x) >> 5;
    int lane = threadIdx.x & 31;
    int nblk = (n + 15) >> 4;
    if (wave >= nblk) return;          // wave-uniform: EXEC stays all-ones for WMMA
    int base = wave << 4;
    int m    = lane & 15;
    int half = lane >> 4;
    int k0   = half * 2;

    int node  = base + m;
    int nodec = node < n ? node : (n - 1);   // clamp loads for tail block
    float s = dinv[nodec]; s = s * s;        // self-loop norm dinv^2

    v2f aA, aB;
    aA.x = a4a[nodec * 4 + k0    ] + s * h1[nodec * 4 + k0    ];
    aA.y = a4a[nodec * 4 + k0 + 1] + s * h1[nodec * 4 + k0 + 1];
    aB.x = a4b[nodec * 4 + k0    ] + s * h2[nodec * 4 + k0    ];
    aB.y = a4b[nodec * 4 + k0 + 1] + s * h2[nodec * 4 + k0 + 1];

    v2f bA = {0.f, 0.f}, bB = {0.f, 0.f};
    if (m < 4) {                       // W is 4x4 row-major [k][nout]; pad cols 4..15 with 0
        bA.x = w1[(k0    ) * 4 + m];
        bA.y = w1[(k0 + 1) * 4 + m];
        bB.x = w2[(k0    ) * 4 + m];
        bB.y = w2[(k0 + 1) * 4 + m];
    }

    v8f c0 = {};
    v8f d1 = __builtin_amdgcn_wmma_f32_16x16x4_f32(false, aA, false, bA, (short)0, c0, false, false);
    v8f d2 = __builtin_amdgcn_wmma_f32_16x16x4_f32(false, aB, false, bB, (short)0, c0, false, false);

    if (m < 4) {                       // only columns 0..3 are valid outputs
        float bias1 = b1[m], bias2 = b2[m];
        #pragma unroll
        for (int v = 0; v < 8; ++v) {
            int nd = base + v + 8 * half;   // row M = v + 8*half
            if (nd < n) {
                y1[nd * 4 + m] = d1[v] + bias1;
                y2[nd * 4 + m] = d2[v] + bias2;
            }
        }
    }
}

// ---------------- final pair reduction: out[2][4] ----------------
__global__ void k_reduce(const float4* __restrict__ y1, const float4* __restrict__ y2,
                         const long long* __restrict__ src, const long long* __restrict__ dst,
                         float* __restrict__ out, int k) {
    __shared__ float sred[8];
    int tid = threadIdx.x;
    if (tid < 8) sred[tid] = 0.0f;
    __syncthreads();
    float acc[8];
    #pragma unroll
    for (int c = 0; c < 8; ++c) acc[c] = 0.0f;
    for (int i = blockIdx.x * blockDim.x + tid; i < k; i += gridDim.x * blockDim.x) {
        int a = (int)src[i], b = (int)dst[i];
        float4 sa1 = y1[a], sb1 = y1[b];
        float4 sa2 = y2[a], sb2 = y2[b];
        acc[0] += sa1.x * sb1.x; acc[1] += sa1.y * sb1.y;
        acc[2] += sa1.z * sb1.z; acc[3] += sa1.w * sb1.w;
        acc[4] += sa2.x * sb2.x; acc[5] += sa2.y * sb2.y;
        acc[6] += sa2.z * sb2.z; acc[7] += sa2.w * sb2.w;
    }
    #pragma unroll
    for (int c = 0; c < 8; ++c) atomicAdd(&sred[c], acc[c]);   // LDS ds_add_f32
    __syncthreads();
    if (tid < 8) atomicAdd(&out[tid], sred[tid]);
}

extern "C" void kernel_launch(void* const* d_in, const int* in_sizes, int n_in,
                              void* d_out, int out_size, void* d_ws, size_t ws_size,
                              hipStream_t stream) {
    (void)n_in; (void)out_size; (void)ws_size;
    const float*     x   = (const float*)d_in[0];
    const long long* ei  = (const long long*)d_in[1];
    const long long* src = (const long long*)d_in[2];
    const long long* dst = (const long long*)d_in[3];
    const float* w1_0 = (const float*)d_in[4];
    const float* b1_0 = (const float*)d_in[5];
    const float* w1_1 = (const float*)d_in[6];
    const float* b1_1 = (const float*)d_in[7];
    const float* w2_0 = (const float*)d_in[8];
    const float* b2_0 = (const float*)d_in[9];
    const float* w2_1 = (const float*)d_in[10];
    const float* b2_1 = (const float*)d_in[11];
    float* out = (float*)d_out;

    const int N = in_sizes[0];
    const int E = in_sizes[1] / 2;
    const int K = in_sizes[2];
    const long long* row = ei;       // edge_index[0]
    const long long* col = ei + E;   // edge_index[1]

    // workspace layout (floats), each slab aligned to 4 floats
    size_t Na = ((size_t)N + 3) & ~(size_t)3;
    float* ws   = (float*)d_ws;
    float* deg  = ws;                 // N   (becomes dinv in place)
    float* aggS = ws + Na;            // N
    float* h1   = ws + 2 * Na;        // 4N
    float* h2   = ws + 6 * Na;        // 4N
    float* a4a  = ws + 10 * Na;       // 4N
    float* a4b  = ws + 14 * Na;       // 4N
    float* y1   = ws + 18 * Na;       // 4N
    float* y2   = ws + 22 * Na;       // 4N

    const int T = 256;
    int gN = (N + T - 1) / T;
    int gE = (E + T - 1) / T;

    k_init<<<gN, T, 0, stream>>>(deg, aggS, (float4*)a4a, (float4*)a4b, out, N);
    k_degree<<<gE, T, 0, stream>>>(col, deg, E);
    k_dinv<<<gN, T, 0, stream>>>(deg, N);
    k_agg1<<<gE, T, 0, stream>>>(row, col, deg, x, aggS, E);
    k_layer1<<<gN, T, 0, stream>>>(aggS, deg, x, w1_0, b1_0, w2_0, b2_0,
                                   (float4*)h1, (float4*)h2, N);
    k_agg2<<<gE, T, 0, stream>>>(row, col, deg, (const float4*)h1, (const float4*)h2,
                                 a4a, a4b, E);
    int nblk  = (N + 15) / 16;            // one wave per 16 nodes
    int gW    = (nblk * 32 + T - 1) / T;
    k_layer2_wmma<<<gW, T, 0, stream>>>(a4a, a4b, h1, h2, deg,
                                        w1_1, b1_1, w2_1, b2_1, y1, y2, N);
    int gR = 2048;
    k_reduce<<<gR, T, 0, stream>>>((const float4*)y1, (const float4*)y2, src, dst, out, K);
}